// StochasticTwoLayerRGCN_80977313399043
// MI455X (gfx1250) — compile-verified
//
#include <hip/hip_runtime.h>
#include <stddef.h>

// ---------------- problem constants (from reference) ----------------
static constexpr int R      = 4;
static constexpr int N_DST1 = 20000;
static constexpr int N_DST2 = 4000;
static constexpr int E1     = 160000;
static constexpr int E2     = 32000;
static constexpr int D_IN   = 256;   // = D_HID
static constexpr int D_HID  = 256;
static constexpr int D_OUT  = 128;

typedef __attribute__((ext_vector_type(2))) float v2f;
typedef __attribute__((ext_vector_type(8))) float v8f;

// ---------------- utility kernels ----------------

__global__ void zero_i32(int* __restrict__ p, int n) {
  int i = blockIdx.x * blockDim.x + threadIdx.x;
  if (i < n) p[i] = 0;
}

// out[n, c] = sum_r b[r, c]   (out flattened [Nrows*C], C power of two)
__global__ void init_bias_sum(const float* __restrict__ b, float* __restrict__ out,
                              int total, int C) {
  int i = blockIdx.x * blockDim.x + threadIdx.x;
  if (i >= total) return;
  int c = i & (C - 1);
  out[i] = b[c] + b[C + c] + b[2 * C + c] + b[3 * C + c];
}

// ---------------- CSR build: histogram -> scan -> bucket fill ----------------

__global__ void hist_dst(const int* __restrict__ dst, int* __restrict__ count, int E) {
  int e = blockIdx.x * blockDim.x + threadIdx.x;
  if (e < E) atomicAdd(&count[dst[e]], 1);
}

// Single-workgroup exclusive scan (n up to a few 10k). blockDim.x must be 1024.
__global__ void exclusive_scan(const int* __restrict__ in, int* __restrict__ out, int n) {
  __shared__ int buf[1024];
  __shared__ int carry;
  if (threadIdx.x == 0) carry = 0;
  __syncthreads();
  for (int base = 0; base < n; base += 1024) {
    int i = base + threadIdx.x;
    int v = (i < n) ? in[i] : 0;
    buf[threadIdx.x] = v;
    __syncthreads();
    // Hillis-Steele inclusive scan in LDS
    for (int off = 1; off < 1024; off <<= 1) {
      int t = (threadIdx.x >= off) ? buf[threadIdx.x - off] : 0;
      __syncthreads();
      buf[threadIdx.x] += t;
      __syncthreads();
    }
    int incl = buf[threadIdx.x];
    int total = buf[1023];
    if (i < n) out[i] = incl - v + carry;
    __syncthreads();
    if (threadIdx.x == 0) carry += total;
    __syncthreads();
  }
}

__global__ void fill_buckets(const int* __restrict__ src, const int* __restrict__ dst,
                             const int* __restrict__ rowptr, int* __restrict__ cursor,
                             int* __restrict__ col, int E) {
  int e = blockIdx.x * blockDim.x + threadIdx.x;
  if (e >= E) return;
  int d = dst[e];
  int pos = atomicAdd(&cursor[d], 1);
  col[rowptr[d] + pos] = src[e];
}

// ---------------- gather-sum + right-normalize (no float atomics) ----------------
// One 64-thread block per destination row; lane owns one float4 chunk of the
// 256-float row. Edge list is uniform per block -> scalar loads.
__global__ void gather_rows(const float* __restrict__ xs,
                            const int* __restrict__ col,
                            const int* __restrict__ rowptr,
                            const int* __restrict__ count,
                            float* __restrict__ aggn) {   // [n_dst x 256], normalized
  const int d = blockIdx.x;
  const int c = threadIdx.x;           // 0..63 float4 chunks
  const int start = rowptr[d];
  const int deg = count[d];
  float4 acc = make_float4(0.f, 0.f, 0.f, 0.f);
  for (int j = 0; j < deg; ++j) {
    int s = col[start + j];            // uniform -> s_load
    if (j + 1 < deg) {
      int sn = col[start + j + 1];
      __builtin_prefetch(xs + (size_t)sn * 256 + c * 4, 0, 3);  // global_prefetch_b8
    }
    float4 v = ((const float4*)(xs + (size_t)s * 256))[c];
    acc.x += v.x; acc.y += v.y; acc.z += v.z; acc.w += v.w;
  }
  const float inv = 1.0f / fmaxf((float)deg, 1.0f);
  acc.x *= inv; acc.y *= inv; acc.z *= inv; acc.w *= inv;
  ((float4*)(aggn + (size_t)d * 256))[c] = acc;
}

// ---------------- dense phase: out[m,n] += A[m,:] @ W[:,n]  (A pre-normalized) --------
// K = 256 fixed. Each wave computes a 32(M) x 64(N) tile: two 16x16x4 WMMA M-tiles
// share every B fetch. blockDim.x = 32*(Nout/64); wave w covers cols [w*64, w*64+64).
// gridDim.x = Mrows/32. No divergence -> EXEC all-1s for WMMA.
__global__ void gemm_acc(const float* __restrict__ A,   // [Mrows x 256]
                         const float* __restrict__ W,   // [256 x Nout], relation-offset
                         float* __restrict__ out,       // [Mrows x Nout]
                         int Nout) {
  const int lane = threadIdx.x & 31;
  const int wid  = threadIdx.x >> 5;
  const int m0   = blockIdx.x * 32;
  const int n0   = wid * 64;
  const int mlo  = lane & 15;           // A: M index / B,C: N index
  const int kg   = (lane >> 4) << 1;    // K-pair base: 0 or 2

  const float* a0 = A + (size_t)(m0 + mlo) * 256 + kg;
  const float* a1 = a0 + (size_t)16 * 256;
  const float* wcol = W + (size_t)kg * Nout + n0 + mlo;

  v8f c00 = {}, c01 = {}, c02 = {}, c03 = {};
  v8f c10 = {}, c11 = {}, c12 = {}, c13 = {};

  #pragma unroll 4
  for (int k0 = 0; k0 < 256; k0 += 4) {
    v2f aA = *(const v2f*)(a0 + k0);
    v2f aB = *(const v2f*)(a1 + k0);
    const float* wp = wcol + (size_t)k0 * Nout;
    v2f b0, b1, b2, b3;
    b0.x = wp[0];  b0.y = wp[Nout];
    b1.x = wp[16]; b1.y = wp[Nout + 16];
    b2.x = wp[32]; b2.y = wp[Nout + 32];
    b3.x = wp[48]; b3.y = wp[Nout + 48];
    c00 = __builtin_amdgcn_wmma_f32_16x16x4_f32(false, aA, false, b0, (short)0, c00, false, false);
    c10 = __builtin_amdgcn_wmma_f32_16x16x4_f32(false, aB, false, b0, (short)0, c10, false, false);
    c01 = __builtin_amdgcn_wmma_f32_16x16x4_f32(false, aA, false, b1, (short)0, c01, false, false);
    c11 = __builtin_amdgcn_wmma_f32_16x16x4_f32(false, aB, false, b1, (short)0, c11, false, false);
    c02 = __builtin_amdgcn_wmma_f32_16x16x4_f32(false, aA, false, b2, (short)0, c02, false, false);
    c12 = __builtin_amdgcn_wmma_f32_16x16x4_f32(false, aB, false, b2, (short)0, c12, false, false);
    c03 = __builtin_amdgcn_wmma_f32_16x16x4_f32(false, aA, false, b3, (short)0, c03, false, false);
    c13 = __builtin_amdgcn_wmma_f32_16x16x4_f32(false, aB, false, b3, (short)0, c13, false, false);
  }

  // C/D layout: VGPR i -> M = mtile + (lane>>4)*8 + i, N = n0 + j*16 + (lane&15)
  const int rbase = m0 + (lane >> 4) * 8;
  float* ob0 = out + (size_t)rbase * Nout + n0 + mlo;
  float* ob1 = ob0 + (size_t)16 * Nout;
  #pragma unroll
  for (int i = 0; i < 8; ++i) {
    float* r0 = ob0 + (size_t)i * Nout;
    float* r1 = ob1 + (size_t)i * Nout;
    r0[0]  += c00[i];  r0[16] += c01[i];  r0[32] += c02[i];  r0[48] += c03[i];
    r1[0]  += c10[i];  r1[16] += c11[i];  r1[32] += c12[i];  r1[48] += c13[i];
  }
}

// ---------------- host-side launch ----------------

static inline int cdiv(int a, int b) { return (a + b - 1) / b; }

extern "C" void kernel_launch(void* const* d_in, const int* in_sizes, int n_in,
                              void* d_out, int out_size, void* d_ws, size_t ws_size,
                              hipStream_t stream) {
  (void)in_sizes; (void)n_in; (void)out_size; (void)ws_size;

  const float* x    = (const float*)d_in[0];
  const int*   src1 = (const int*)d_in[1];
  const int*   dst1 = (const int*)d_in[2];
  const int*   src2 = (const int*)d_in[3];
  const int*   dst2 = (const int*)d_in[4];
  const float* W1   = (const float*)d_in[5];
  const float* b1   = (const float*)d_in[6];
  const float* W2   = (const float*)d_in[7];
  const float* b2   = (const float*)d_in[8];
  float* out = (float*)d_out;   // [N_DST2, D_OUT]

  // workspace layout
  float* ws    = (float*)d_ws;
  float* aggn1 = ws;                                  // N_DST1 * 256 (normalized agg)
  float* h1    = aggn1 + (size_t)N_DST1 * 256;        // N_DST1 * 256
  float* aggn2 = h1 + (size_t)N_DST1 * 256;           // N_DST2 * 256
  int*   ibuf  = (int*)(aggn2 + (size_t)N_DST2 * 256);
  int* count  = ibuf;                                 // max(N_DST1, N_DST2)
  int* rowptr = count + N_DST1;
  int* cursor = rowptr + N_DST1;
  int* col    = cursor + N_DST1;                      // max(E1, E2)

  // ---- layer 1: h1 = sum_r ( (A_r x / deg_r) @ W1_r + b1_r ) ----
  init_bias_sum<<<cdiv(N_DST1 * D_HID, 256), 256, 0, stream>>>(b1, h1, N_DST1 * D_HID, D_HID);

  for (int r = 0; r < R; ++r) {
    const int* s = src1 + (size_t)r * E1;
    const int* d = dst1 + (size_t)r * E1;
    zero_i32<<<cdiv(N_DST1, 256), 256, 0, stream>>>(count, N_DST1);
    zero_i32<<<cdiv(N_DST1, 256), 256, 0, stream>>>(cursor, N_DST1);
    hist_dst<<<cdiv(E1, 256), 256, 0, stream>>>(d, count, E1);
    exclusive_scan<<<1, 1024, 0, stream>>>(count, rowptr, N_DST1);
    fill_buckets<<<cdiv(E1, 256), 256, 0, stream>>>(s, d, rowptr, cursor, col, E1);
    gather_rows<<<N_DST1, 64, 0, stream>>>(x, col, rowptr, count, aggn1);
    gemm_acc<<<N_DST1 / 32, 32 * (D_HID / 64), 0, stream>>>(
        aggn1, W1 + (size_t)r * D_IN * D_HID, h1, D_HID);
  }

  // ---- layer 2: out = sum_r ( (A_r h1 / deg_r) @ W2_r + b2_r ) ----
  init_bias_sum<<<cdiv(N_DST2 * D_OUT, 256), 256, 0, stream>>>(b2, out, N_DST2 * D_OUT, D_OUT);

  for (int r = 0; r < R; ++r) {
    const int* s = src2 + (size_t)r * E2;
    const int* d = dst2 + (size_t)r * E2;
    zero_i32<<<cdiv(N_DST2, 256), 256, 0, stream>>>(count, N_DST2);
    zero_i32<<<cdiv(N_DST2, 256), 256, 0, stream>>>(cursor, N_DST2);
    hist_dst<<<cdiv(E2, 256), 256, 0, stream>>>(d, count, E2);
    exclusive_scan<<<1, 1024, 0, stream>>>(count, rowptr, N_DST2);
    fill_buckets<<<cdiv(E2, 256), 256, 0, stream>>>(s, d, rowptr, cursor, col, E2);
    gather_rows<<<N_DST2, 64, 0, stream>>>(h1, col, rowptr, count, aggn2);
    gemm_acc<<<N_DST2 / 32, 32 * (D_OUT / 64), 0, stream>>>(
        aggn2, W2 + (size_t)r * D_HID * D_OUT, out, D_OUT);
  }
}